// CNN_61323543052332
// MI455X (gfx1250) — compile-verified
//
#include <hip/hip_runtime.h>
#include <hip/hip_bf16.h>
#include <stdint.h>

#define GRID_PTS 250
#define OFFSET_PTS 125
#define ROW_D 42
#define ROW_CELLS (GRID_PTS * GRID_PTS)   // 62500 floats per row

// ---------------------------------------------------------------------------
// Kernel 1: zero-fill the 2.048 GB output via the CDNA5 async LDS->global
// store path (GLOBAL_STORE_ASYNC_FROM_LDS_B128, ASYNCcnt-tracked, up to 63
// in flight; hardware stalls issue on counter saturation).
//
// Tiling: each block owns contiguous 32 KB tiles (256 lanes x 16 B x 8).
// Thread t stores chunk (tile*2048 + t) and covers the 8 sub-rows with the
// instruction's 24-bit immediate offset (0..28672), so ONE 64-bit address
// computation feeds EIGHT async B128 stores: ~0.6 VALU instructions per
// 512 B/wave store, leaving the wave pinned on ASYNCcnt back-pressure.
// ---------------------------------------------------------------------------
__global__ void CNN_zero_async_kernel(float* __restrict__ out, long long n4 /*16B chunks*/,
                                      long long ntail /*leftover floats*/) {
    __shared__ __align__(16) float zbuf[256 * 4];   // 16 B per thread
    const int tid = threadIdx.x;

    // Fill this lane's 16B LDS chunk with zeros.
    float4 z = make_float4(0.f, 0.f, 0.f, 0.f);
    *reinterpret_cast<float4*>(&zbuf[tid * 4]) = z;
    __syncthreads();   // DS writes visible before the async engine reads LDS

    // Low 32 bits of the flat shared-memory address == LDS byte offset
    // (LDS aperture truncates to addr[31:0]).
    const uint32_t lds_off = (uint32_t)(uintptr_t)(&zbuf[tid * 4]);

    const long long CHUNKS_PER_TILE = 256LL * 8;            // 2048 chunks = 32 KB
    const long long ntiles = n4 / CHUNKS_PER_TILE;

    for (long long tile = blockIdx.x; tile < ntiles; tile += gridDim.x) {
        uint64_t ga = (uint64_t)(uintptr_t)out +
                      (uint64_t)((tile * CHUNKS_PER_TILE + tid) << 4);
        asm volatile(
            "global_store_async_from_lds_b128 %0, %1, off\n\t"
            "global_store_async_from_lds_b128 %0, %1, off offset:4096\n\t"
            "global_store_async_from_lds_b128 %0, %1, off offset:8192\n\t"
            "global_store_async_from_lds_b128 %0, %1, off offset:12288\n\t"
            "global_store_async_from_lds_b128 %0, %1, off offset:16384\n\t"
            "global_store_async_from_lds_b128 %0, %1, off offset:20480\n\t"
            "global_store_async_from_lds_b128 %0, %1, off offset:24576\n\t"
            "global_store_async_from_lds_b128 %0, %1, off offset:28672\n\t"
            :
            : "v"(ga), "v"(lds_off)
            : "memory");
    }

    // Remainder chunks (n4 = 128M is exactly tile-divisible, but stay robust).
    const long long rem_start = ntiles * CHUNKS_PER_TILE;
    for (long long idx = rem_start + (long long)blockIdx.x * blockDim.x + tid;
         idx < n4;
         idx += (long long)gridDim.x * blockDim.x) {
        uint64_t ga = (uint64_t)(uintptr_t)(out + idx * 4);
        asm volatile("global_store_async_from_lds_b128 %0, %1, off"
                     :
                     : "v"(ga), "v"(lds_off)
                     : "memory");
    }

    // Drain this wave's async stores explicitly (S_ENDPGM would also drain).
    asm volatile("s_wait_asynccnt 0x0" ::: "memory");

    // Scalar tail floats (none here: out_size is 16B-divisible).
    if (blockIdx.x == 0 && (long long)tid < ntail) {
        out[n4 * 4 + tid] = 0.0f;
    }
}

// ---------------------------------------------------------------------------
// Kernel 2: per-row normalize + quantize + scatter. One thread per row:
// 42 loads, sum of squares, sqrt, then 21 (a,b) index pairs -> store 1.0f
// with bounds check (reference uses mode="drop"; duplicates are idempotent).
// Runs after kernel 1 on the same stream, so zeros are globally visible
// before the ones land.
// ---------------------------------------------------------------------------
__global__ void CNN_scatter_kernel(const float* __restrict__ x,
                                   float* __restrict__ out, int rows) {
    int r = blockIdx.x * blockDim.x + threadIdx.x;
    if (r >= rows) return;

    const float* xr = x + (long long)r * ROW_D;

    float v[ROW_D];
    float s = 0.0f;
#pragma unroll
    for (int i = 0; i < ROW_D; ++i) {
        v[i] = xr[i];
        s += v[i] * v[i];
    }
    float norm = fmaxf(sqrtf(s), 1e-12f);   // F.normalize eps clamp

    float* gr = out + (long long)r * ROW_CELLS;
#pragma unroll
    for (int k = 0; k < ROW_D / 2; ++k) {
        // Match reference op order: (x / norm) * 250, round-half-even (rintf).
        int a = (int)rintf((v[2 * k]     / norm) * (float)GRID_PTS) + OFFSET_PTS;
        int b = (int)rintf((v[2 * k + 1] / norm) * (float)GRID_PTS) + OFFSET_PTS;
        if ((unsigned)a < (unsigned)GRID_PTS && (unsigned)b < (unsigned)GRID_PTS) {
            gr[a * GRID_PTS + b] = 1.0f;
        }
    }
}

extern "C" void kernel_launch(void* const* d_in, const int* in_sizes, int n_in,
                              void* d_out, int out_size, void* d_ws, size_t ws_size,
                              hipStream_t stream) {
    const float* x = (const float*)d_in[0];
    float* out = (float*)d_out;

    const int rows = in_sizes[0] / ROW_D;               // 8192
    const long long n  = (long long)out_size;           // 512,000,000 floats
    const long long n4 = n / 4;                         // 128M 16B chunks
    const long long ntail = n % 4;

    // Zero-fill: 62,500 tiles of 32 KB; 16384 blocks grid-stride over them
    // (~4 tiles/block, 32 async B128 stores per lane).
    CNN_zero_async_kernel<<<16384, 256, 0, stream>>>(out, n4, ntail);

    // Scatter: one thread per row.
    const int threads = 256;
    const int blocks = (rows + threads - 1) / threads;
    CNN_scatter_kernel<<<blocks, threads, 0, stream>>>(x, out, rows);
}